// YOLOLoss_73186242724567
// MI455X (gfx1250) — compile-verified
//
#include <hip/hip_runtime.h>
#include <hip/hip_bf16.h>

// ---------------------------------------------------------------------------
// YOLO loss on gfx1250.
//   d_in[0]=pred0 (16,13,13,3,85) f32   d_in[1]=tgt0
//   d_in[2]=pred1 (16,26,26,3,85) f32   d_in[3]=tgt1
//   d_in[4]=pred2 (16,52,52,3,85) f32   d_in[5]=tgt2
//   d_in[6]=anchors (9,2) f32
//   d_out[0] = scalar loss (f32)
//
// Workspace layout (floats):
//   [0 .. 48*132)        : 48 box slots (layer*16+b), each 32 boxes*4 + count
//   [6336 .. 6336+672)   : per-block partial sums (l0:32, l1:128, l2:512)
// ---------------------------------------------------------------------------

#define NB       16
#define NA       3
#define NC       85
#define MAXBOX   32
#define SLOT_F   132            // 32*4 boxes + 1 count (+3 pad)
#define BOX_REGION (48 * SLOT_F)
#define NPART    (32 + 128 + 512)

typedef __attribute__((ext_vector_type(2))) float v2f;
typedef __attribute__((ext_vector_type(8))) float v8f;

// Wave32 sum via V_WMMA_F32_16X16X4_F32.
// A (16x4 f32): lanes 0-15 hold rows M=lane with K=0 (vgpr0), K=1 (vgpr1);
// lanes 16-31 hold the same rows with K=2,3.  Put the lane partial in vgpr0,
// zero in vgpr1 -> A[m,0]=s[m], A[m,2]=s[m+16].  B = ones  =>
// D[m,n] = s[m] + s[m+16].  D vgpr v holds row v (lanes 0-15) / row v+8
// (lanes 16-31), so summing the 8 accumulator VGPRs gives rows 0..7 on the
// low half-wave and rows 8..15 on the high half-wave; one xor-16 shuffle
// finishes the full 32-lane sum (all in f32).
__device__ __forceinline__ float wave_sum_wmma(float s) {
    v2f a; a[0] = s;    a[1] = 0.0f;
    v2f o; o[0] = 1.0f; o[1] = 1.0f;
    v8f c = {};
    v8f d = __builtin_amdgcn_wmma_f32_16x16x4_f32(
        /*neg_a=*/false, a, /*neg_b=*/false, o,
        /*c_mod=*/(short)0, c, /*reuse_a=*/false, /*reuse_b=*/false);
    float t = ((d[0] + d[1]) + (d[2] + d[3])) + ((d[4] + d[5]) + (d[6] + d[7]));
    t += __shfl_xor(t, 16, 32);
    return t;   // every lane holds the wave total
}

// Block sum (result valid on thread 0).  Must be called by all threads.
__device__ __forceinline__ float block_sum(float s, float* smem8) {
    float w = wave_sum_wmma(s);
    int wave = threadIdx.x >> 5;
    if ((threadIdx.x & 31) == 0) smem8[wave] = w;
    __syncthreads();
    float tot = 0.0f;
    if (threadIdx.x == 0) {
        int nw = blockDim.x >> 5;
        for (int i = 0; i < nw; ++i) tot += smem8[i];   // fixed order
    }
    return tot;
}

// ---------------------------------------------------------------------------
// Stage 1: gather positive GT boxes (first <=32 in flat (y,x,a) order, which
// matches lax.top_k's stable ordering on {0,1} obj values).
// One block of 32 threads per image.
// ---------------------------------------------------------------------------
__global__ void yolo_gather_boxes(const float* __restrict__ tgt,
                                  float* __restrict__ ws,
                                  int ncells, int slotBase) {
    int b = blockIdx.x;
    const float* t = tgt + (size_t)b * ncells * NC;
    float* slot = ws + (size_t)(slotBase + b) * SLOT_F;
    int lane = threadIdx.x;
    int count = 0;
    for (int base = 0; base < ncells; base += 32) {
        int cell = base + lane;
        bool pos = false;
        float bx = 0.f, by = 0.f, bw = 0.f, bh = 0.f;
        if (cell < ncells) {
            const float* tc = t + (size_t)cell * NC;
            if (tc[4] > 0.0f) {
                pos = true;
                bx = tc[0]; by = tc[1]; bw = tc[2]; bh = tc[3];
            }
        }
        unsigned long long m = __ballot(pos);
        if (pos) {
            int idx = count + __popcll(m & ((1ull << lane) - 1ull));
            if (idx < MAXBOX) {
                slot[idx * 4 + 0] = bx;
                slot[idx * 4 + 1] = by;
                slot[idx * 4 + 2] = bw;
                slot[idx * 4 + 3] = bh;
            }
        }
        count += (int)__popcll(m);
        if (count >= MAXBOX) break;
    }
    if (lane == 0) slot[MAXBOX * 4] = (float)(count < MAXBOX ? count : MAXBOX);
}

// ---------------------------------------------------------------------------
// Stage 2: per-cell loss.  grid = (blocksPerImg, 16 images), block = 256.
// Fast path (obj==0): only p[0..4], t[4] touched -> ~6 of 170 floats/cell.
// ---------------------------------------------------------------------------
__global__ void yolo_loss_kernel(const float* __restrict__ pred,
                                 const float* __restrict__ tgt,
                                 const float* __restrict__ anchors,
                                 const float* __restrict__ ws,
                                 float* __restrict__ partials,
                                 int G, int ancBase, int slotBase,
                                 int partialBase, int blocksPerImg) {
    __shared__ float sbox[MAXBOX * 4];
    __shared__ int   scount;
    __shared__ float swsum[8];

    int b = blockIdx.y;
    const float* slot = ws + (size_t)(slotBase + b) * SLOT_F;
    if (threadIdx.x < MAXBOX * 4) sbox[threadIdx.x] = slot[threadIdx.x];
    if (threadIdx.x == 0) scount = (int)slot[MAXBOX * 4];
    __syncthreads();

    int ncells = G * G * NA;
    int cell = blockIdx.x * blockDim.x + threadIdx.x;
    float s = 0.0f;

    if (cell < ncells) {
        const float* p = pred + ((size_t)b * ncells + cell) * NC;
        const float* t = tgt  + ((size_t)b * ncells + cell) * NC;
        int a  = cell % NA;
        int xy = cell / NA;
        int gx = xy % G;
        int gy = xy / G;
        float ax = anchors[(ancBase + a) * 2 + 0];
        float ay = anchors[(ancBase + a) * 2 + 1];
        float invG = 1.0f / (float)G;

        float p0 = p[0], p1 = p[1], p2 = p[2], p3 = p[3], p4 = p[4];
        float obj = t[4];

        // predicted box (normalized)
        float px = (p0 + (float)gx) * invG;
        float py = (p1 + (float)gy) * invG;
        float pw = expf(p2) * ax * (1.0f / 416.0f);
        float ph = expf(p3) * ay * (1.0f / 416.0f);
        float parea = pw * ph;

        // ignore mask: max IoU vs gathered GT boxes
        float maxiou = 0.0f;
        int cnt = scount;
        for (int k = 0; k < cnt; ++k) {
            float tx = sbox[k * 4 + 0], ty = sbox[k * 4 + 1];
            float tw = sbox[k * 4 + 2], th = sbox[k * 4 + 3];
            float minx = fmaxf(px - pw * 0.5f, tx - tw * 0.5f);
            float maxx = fminf(px + pw * 0.5f, tx + tw * 0.5f);
            float miny = fmaxf(py - ph * 0.5f, ty - th * 0.5f);
            float maxy = fminf(py + ph * 0.5f, ty + th * 0.5f);
            float iw = fmaxf(maxx - minx, 0.0f);
            float ih = fmaxf(maxy - miny, 0.0f);
            float inter = iw * ih;
            float iou = inter / (parea + tw * th - inter);
            maxiou = fmaxf(maxiou, iou);
        }
        float ignore = (maxiou < 0.5f) ? 1.0f : 0.0f;

        // conf loss: BCE(p4, obj) * (obj + (1-obj)*ignore)
        float conf_w = obj + (1.0f - obj) * ignore;
        s = -(obj * logf(p4) + (1.0f - obj) * log1pf(-p4)) * conf_w;

        if (obj > 0.0f) {   // obj is exactly 1.0 here
            float t0 = t[0], t1 = t[1], t2 = t[2], t3 = t[3];
            float scale = 2.0f - t2 * t3;
            float rtx = t0 * (float)G - (float)gx;
            float rty = t1 * (float)G - (float)gy;
            float twr = t2 * 416.0f / ax;
            float thr = t3 * 416.0f / ay;
            float rtw = (twr != 0.0f) ? logf(twr) : 0.0f;
            float rth = (thr != 0.0f) ? logf(thr) : 0.0f;

            float bce_x = -(rtx * logf(p0) + (1.0f - rtx) * log1pf(-p0));
            float bce_y = -(rty * logf(p1) + (1.0f - rty) * log1pf(-p1));
            s += (bce_x + bce_y) * scale;

            float dw = p2 - rtw, dh = p3 - rth;
            s += 0.5f * (dw * dw + dh * dh) * scale;

            float cls = 0.0f;
            for (int c = 5; c < NC; ++c) {
                float pc = p[c], tc = t[c];
                cls += -(tc * logf(pc) + (1.0f - tc) * log1pf(-pc));
            }
            s += cls;
        }
    }

    // all 256 threads reconverged here -> EXEC all ones for WMMA reduce
    float tot = block_sum(s, swsum);
    if (threadIdx.x == 0)
        partials[partialBase + b * blocksPerImg + blockIdx.x] = tot;
}

// ---------------------------------------------------------------------------
// Stage 3: deterministic fixed-order sum of the 672 block partials, /16.
// ---------------------------------------------------------------------------
__global__ void yolo_finalize(const float* __restrict__ partials,
                              float* __restrict__ out, int n) {
    __shared__ float swsum[8];
    float s = 0.0f;
    for (int i = threadIdx.x; i < n; i += blockDim.x) s += partials[i];
    float tot = block_sum(s, swsum);
    if (threadIdx.x == 0) out[0] = tot * (1.0f / 16.0f);
}

// ---------------------------------------------------------------------------
extern "C" void kernel_launch(void* const* d_in, const int* in_sizes, int n_in,
                              void* d_out, int out_size, void* d_ws, size_t ws_size,
                              hipStream_t stream) {
    const float* pred[3] = { (const float*)d_in[0], (const float*)d_in[2], (const float*)d_in[4] };
    const float* tgtp[3] = { (const float*)d_in[1], (const float*)d_in[3], (const float*)d_in[5] };
    const float* anchors = (const float*)d_in[6];
    float* out = (float*)d_out;

    float* ws = (float*)d_ws;
    float* boxes = ws;                       // 48 * 132 floats
    float* partials = ws + BOX_REGION;       // 672 floats

    const int   G[3]       = { 13, 26, 52 };
    const int   ancBase[3] = { 6, 3, 0 };    // ANCHOR_MASK rows

    int partialBase = 0;
    for (int l = 0; l < 3; ++l) {
        int ncells = G[l] * G[l] * NA;
        yolo_gather_boxes<<<NB, 32, 0, stream>>>(tgtp[l], boxes, ncells, l * NB);
    }
    for (int l = 0; l < 3; ++l) {
        int ncells = G[l] * G[l] * NA;
        int bpi = (ncells + 255) / 256;
        dim3 grid(bpi, NB);
        yolo_loss_kernel<<<grid, 256, 0, stream>>>(pred[l], tgtp[l], anchors,
                                                   boxes, partials,
                                                   G[l], ancBase[l], l * NB,
                                                   partialBase, bpi);
        partialBase += bpi * NB;
    }
    yolo_finalize<<<1, 256, 0, stream>>>(partials, out, NPART);
}